// M_H_40415642255477
// MI455X (gfx1250) — compile-verified
//
#include <hip/hip_runtime.h>
#include <hip/hip_bf16.h>

// Problem: B=4, P=128, N=2048, K=64, I=32
//  out[b,n,i,k] = ifft_k( sum_l w[i,j,l] * (FFT_P f_h)[b,i,n,l] )
// Folded form: out[b,n,i,k] = sum_l Wt[i,k,l] * F[b,i,n,l]
//   F  = truncated DFT over P (real input -> cos / -sin real GEMMs)
//   Wt = (1/64) sum_j w[i,j,l] e^{+2*pi*i*j*k/64}   (iFFT folded into weights)
//
// Workspace layout (floats), total ~135.3 MB:
//   [0      .. 4096)      Dr   cos table   (32 x 128)
//   [4096   .. 8192)      Di  -sin table   (32 x 128)   (Dr||Di contiguous: one TDM copy)
//   [8192   .. 139264)    WtR  folded weights real, stored [i][l][k]
//   [139264 .. 270336)    WtI  folded weights imag, stored [i][l][k]
//   [270336 .. +16777216) Fr   (B,I,N,K) real plane
//   [...    .. +16777216) Fi   (B,I,N,K) imag plane

typedef float        v2f __attribute__((ext_vector_type(2)));
typedef float        v8f __attribute__((ext_vector_type(8)));
typedef unsigned int v4u __attribute__((ext_vector_type(4)));
typedef int          v4i __attribute__((ext_vector_type(4)));
typedef int          v8i __attribute__((ext_vector_type(8)));

#define TWO_PI 6.283185307179586f

static __device__ __forceinline__ v8f wmma_f32(v2f a, v2f b, v8f c) {
  // V_WMMA_F32_16X16X4_F32 : D = A(16x4) * B(4x16) + C(16x16)
  return __builtin_amdgcn_wmma_f32_16x16x4_f32(
      /*neg_a=*/false, a, /*neg_b=*/false, b,
      /*c_mod=*/(short)0, c, /*reuse_a=*/false, /*reuse_b=*/false);
}

// Issue one TDM 2-D tile load (global -> LDS), 4-byte elements.
// D# per CDNA5 ISA ch.8: group0 = {count, lds_addr, global_addr, type=2},
// group1 = {data_size/pad flags, tensor dims, tile dims, dim0 stride}.
static __device__ __forceinline__ void tdm_load_2d(
    const void* gaddr, unsigned lds_byte_off, unsigned flags_w0,
    unsigned tdim0, unsigned tdim1, unsigned tile0, unsigned tile1,
    unsigned long long stride0_elems) {
  unsigned long long ga = (unsigned long long)gaddr;
  v4u g0;
  g0[0] = 1u;                                            // count=1, user mode
  g0[1] = lds_byte_off;                                  // lds_addr (bytes)
  g0[2] = (unsigned)(ga & 0xffffffffu);                  // global_addr[31:0]
  g0[3] = (unsigned)((ga >> 32) & 0x01ffffffu)           // global_addr[56:32]
          | (2u << 30);                                  // type = 2 ("image")
  v8i g1;
  g1[0] = (int)flags_w0;                                 // wg_mask=0 | data_size | pad cfg
  g1[1] = (int)((tdim0 & 0xffffu) << 16);                // atomic_addr=0 | tensor_dim0.lo16
  g1[2] = (int)(((tdim0 >> 16) & 0xffffu) | ((tdim1 & 0xffffu) << 16));
  g1[3] = (int)(((tdim1 >> 16) & 0xffffu) | ((tile0 & 0xffffu) << 16));
  g1[4] = (int)(tile1 & 0xffffu);                        // tile_dim1 | tile_dim2=0
  g1[5] = (int)(unsigned)(stride0_elems & 0xffffffffu);  // tensor_dim0_stride.lo32
  g1[6] = (int)(unsigned)((stride0_elems >> 32) & 0xffffu);
  g1[7] = 0;
  v4i z4 = {0, 0, 0, 0};                                 // 2-D tensor: groups 2/3 unused
#if __clang_major__ >= 23
  v8i z8 = {0, 0, 0, 0, 0, 0, 0, 0};
  __builtin_amdgcn_tensor_load_to_lds(g0, g1, z4, z4, z8, 0);
#else
  __builtin_amdgcn_tensor_load_to_lds(g0, g1, z4, z4, 0);
#endif
}

// ---------------------------------------------------------------- k1: DFT tables
__global__ __launch_bounds__(256) void build_dft_tables(float* __restrict__ Dr,
                                                        float* __restrict__ Di) {
  int t = blockIdx.x * blockDim.x + threadIdx.x;   // 0 .. 4095
  if (t >= 32 * 128) return;
  int m = t >> 7, p = t & 127;
  int ph = (m * p) & 127;                          // e^{-2pi i mp/128}, period 128
  float th = (TWO_PI / 128.0f) * (float)ph;
  Dr[t] = __cosf(th);
  Di[t] = -__sinf(th);
}

// -------------------------------------------------- k2: fold iFFT into weights
// Wt[i][l][k] = (1/64) sum_j (wr + i*wi)[i,j,l] * e^{+2pi i jk/64}
__global__ __launch_bounds__(256) void build_folded_weights(
    const float* __restrict__ w,      // (I,K,K) complex64 interleaved
    float* __restrict__ WtR, float* __restrict__ WtI) {
  int t = blockIdx.x * blockDim.x + threadIdx.x;   // 0 .. 131071
  int i = t >> 12;
  int l = (t >> 6) & 63;
  int k = t & 63;
  float sr = 0.f, si = 0.f;
  for (int j = 0; j < 64; ++j) {
    int ph = (j * k) & 63;                         // period 64
    float th = (TWO_PI / 64.0f) * (float)ph;
    float c = __cosf(th), s = __sinf(th);
    const float* wp = w + ((((i << 6) + j) << 6) + l) * 2;
    float wr = wp[0], wi = wp[1];
    sr += wr * c - wi * s;
    si += wr * s + wi * c;
  }
  WtR[t] = sr * (1.0f / 64.0f);
  WtI[t] = si * (1.0f / 64.0f);
}

// ------------------------------------------------ k3: truncated DFT as WMMA GEMM
// Per (b,n):  F(32 modes x 64 l) = D(32x128) * f(128x64), real & imag planes.
// f tile and twiddle tables staged in LDS by the Tensor Data Mover; the f tile is
// pad-on-load to an 80-float row stride so the two lane-halves of a B fragment hit
// disjoint LDS bank groups. Block = 256 threads = 8 waves;
// wave bits: [0]=plane, [1]=mode tile, [2]=l half.
__global__ __launch_bounds__(256) void dft_gemm(
    const float* __restrict__ f,      // (B,P,N,K)
    const float* __restrict__ Dtab,   // Dr||Di, 8192 floats
    float* __restrict__ Fr, float* __restrict__ Fi) {
  __shared__ float sD[8192];          // Dr [0..4096), Di [4096..8192)
  __shared__ float sF[128 * 80];      // f tile, 80-float padded rows (40 KB)

  int bn   = blockIdx.x;              // 0 .. 8191
  int b    = bn >> 11;
  int n    = bn & 2047;
  int lane = threadIdx.x & 31;
  int wave = threadIdx.x >> 5;
  int laneLo = lane & 15, laneHi = lane >> 4;

  if (wave == 0) {
    // Twiddle tables: flat 8192-float copy, no padding. data_size=4B (code 2).
    tdm_load_2d(Dtab, (unsigned)(unsigned long long)(void*)sD,
                /*flags=*/(2u << 16),
                /*tdim0=*/8192, /*tdim1=*/1, /*tile0=*/8192, /*tile1=*/1,
                /*stride0=*/8192ull);
    // f tile: 128 rows x 64 cols, row stride N*K elements in memory;
    // pad_enable=1, pad_interval=5 (every 64 DWORDs), pad_amount=15 (16 DWORDs)
    // -> LDS row stride 80 floats.
    tdm_load_2d(f + (unsigned long long)b * (128ull * 2048 * 64) + (unsigned)n * 64,
                (unsigned)(unsigned long long)(void*)sF,
                /*flags=*/(2u << 16) | (1u << 20) | (5u << 22) | (15u << 25),
                /*tdim0=*/64, /*tdim1=*/128, /*tile0=*/64, /*tile1=*/128,
                /*stride0=*/2048ull * 64ull);
    __builtin_amdgcn_s_wait_tensorcnt(0);
  }
  __syncthreads();

  int plane = wave & 1;
  int m0    = ((wave >> 1) & 1) * 16;            // mode tile base
  int l0    = ((wave >> 2) & 1) * 32 + laneLo;   // this wave's first l column

  const float* Arow = sD + plane * 4096 + (m0 + laneLo) * 128;  // A row m0+laneLo of Dr or Di

  v8f acc0 = {0.f,0.f,0.f,0.f,0.f,0.f,0.f,0.f};
  v8f acc1 = {0.f,0.f,0.f,0.f,0.f,0.f,0.f,0.f};

  for (int kk = 0; kk < 128; kk += 4) {
    int kA = kk + 2 * laneHi;                    // lanes 0-15: K{kk,kk+1}; 16-31: K{kk+2,kk+3}
    v2f a = *(const v2f*)(Arow + kA);            // ds_load_b64, distinct address per lane

    v2f b0, b1;                                  // B[K = kA + j][N = laneLo(+16)]
    b0.x = sF[kA * 80 + l0];
    b0.y = sF[(kA + 1) * 80 + l0];
    b1.x = sF[kA * 80 + l0 + 16];
    b1.y = sF[(kA + 1) * 80 + l0 + 16];

    acc0 = wmma_f32(a, b0, acc0);
    acc1 = wmma_f32(a, b1, acc1);
  }

  // C/D layout: vgpr r, lane -> M = r + 8*laneHi, N = laneLo
  for (int r = 0; r < 8; ++r) {
    int m = m0 + r + 8 * laneHi;
    float* dst = (plane ? Fi : Fr) + (b * 32 + m) * (2048 * 64) + n * 64 + l0;
    dst[0]  = acc0[r];
    dst[16] = acc1[r];
  }
}

// --------------------------------- k4: per-mode complex contraction + folded iFFT
// Per (b,i): Out(2048 n x 64 k) = F_i(n x l) * Wt_i(l x k), complex.
// Block = 128 threads = 4 waves; block handles one 16-row n tile; wave = k tile.
__global__ __launch_bounds__(128) void mode_gemm(
    const float* __restrict__ Fr, const float* __restrict__ Fi,
    const float* __restrict__ WtR, const float* __restrict__ WtI,
    float* __restrict__ out) {
  int blk  = blockIdx.x;              // b*32*128 + i*128 + ntile
  int nt   = blk & 127;
  int bi   = blk >> 7;
  int i    = bi & 31;
  int b    = bi >> 5;
  int lane = threadIdx.x & 31;
  int kt   = threadIdx.x >> 5;        // k tile 0..3
  int laneLo = lane & 15, laneHi = lane >> 4;
  int n0 = nt * 16;

  const float* FrRow = Fr + ((b * 32 + i) * 2048 + n0 + laneLo) * 64;
  const float* FiRow = Fi + ((b * 32 + i) * 2048 + n0 + laneLo) * 64;
  const float* Wr = WtR + i * 4096 + kt * 16 + laneLo;  // + l*64
  const float* Wi = WtI + i * 4096 + kt * 16 + laneLo;

  v8f oR = {0.f,0.f,0.f,0.f,0.f,0.f,0.f,0.f};
  v8f oI = {0.f,0.f,0.f,0.f,0.f,0.f,0.f,0.f};

  for (int ll = 0; ll < 64; ll += 4) {
    int lA = ll + 2 * laneHi;
    v2f ar = *(const v2f*)(FrRow + lA);            // A frag, real plane
    v2f ai = *(const v2f*)(FiRow + lA);            // A frag, imag plane
    if (ll + 4 < 64) {
      __builtin_prefetch(FrRow + lA + 4, 0, 3);    // global_prefetch_b8
      __builtin_prefetch(FiRow + lA + 4, 0, 3);
    }
    v2f br, bi2, bin;
    br.x  = Wr[lA * 64];  br.y  = Wr[(lA + 1) * 64];
    bi2.x = Wi[lA * 64];  bi2.y = Wi[(lA + 1) * 64];
    bin = -bi2;  // f32 WMMA has no A/B NEG (ISA: NEG={CNeg,0,0}) -> negate in VALU

    oR = wmma_f32(ar, br,  oR);   // + Re*Re
    oR = wmma_f32(ai, bin, oR);   // - Im*Im
    oI = wmma_f32(ar, bi2, oI);   // + Re*Im
    oI = wmma_f32(ai, br,  oI);   // + Im*Re
  }

  // out (b, n, i, k) complex64 interleaved; 16 lanes x 8B = 128B contiguous rows
  int kout = kt * 16 + laneLo;
  for (int r = 0; r < 8; ++r) {
    int n = n0 + r + 8 * laneHi;
    float* o = out + ((((b * 2048 + n) * 32 + i) * 64 + kout) << 1);
    v2f val; val.x = oR[r]; val.y = oI[r];
    *(v2f*)o = val;
  }
}

extern "C" void kernel_launch(void* const* d_in, const int* in_sizes, int n_in,
                              void* d_out, int out_size, void* d_ws, size_t ws_size,
                              hipStream_t stream) {
  const float* f_h = (const float*)d_in[0];   // (4,128,2048,64) float32
  const float* w   = (const float*)d_in[1];   // (32,64,64) complex64 interleaved
  float* out = (float*)d_out;                 // (4,2048,32,64) complex64 interleaved
  float* ws  = (float*)d_ws;                  // needs ~135.3 MB

  float* Dr  = ws;                            // Dr||Di contiguous (single TDM copy)
  float* Di  = ws + 4096;
  float* WtR = ws + 8192;
  float* WtI = WtR + 131072;
  float* Fr  = ws + 270336;
  float* Fi  = Fr + 16777216;

  build_dft_tables<<<16, 256, 0, stream>>>(Dr, Di);
  build_folded_weights<<<512, 256, 0, stream>>>(w, WtR, WtI);
  dft_gemm<<<4 * 2048, 256, 0, stream>>>(f_h, Dr, Fr, Fi);
  mode_gemm<<<4 * 32 * 128, 128, 0, stream>>>(Fr, Fi, WtR, WtI, out);
}